// ChainCRFallenNLP_53334903881896
// MI455X (gfx1250) — compile-verified
//
#include <hip/hip_runtime.h>
#include <math.h>

#define N_TAGS 48
#define ROOT_T 46
#define END_T  47
#define BATCH  512
#define MAXT   256

typedef __attribute__((ext_vector_type(2))) float v2f;
typedef __attribute__((ext_vector_type(8))) float v8f;

// ---------------------------------------------------------------------------
// Forward recursion. One wave (32 lanes) owns 16 batch columns.
// State alpha_T[j, b] kept in WMMA C-layout:
//   reg k = frag*8 + v  <->  tag j = frag*16 + 8*h + v,  batch col = lane&15
//   (h = lane>>4). 48 tags -> 3 fragments of 8 VGPRs = 24 regs.
// Constant A = exp(lt)^T held in WMMA A-layout:
//   A[frag][c] : lane row M = lane&15 -> j = frag*16 + (lane&15),
//                K (tag i) = 4c + 2h + {0,1} across the v2f pair.
// ---------------------------------------------------------------------------
__global__ __launch_bounds__(32, 1) void crf_forward(
    const float* __restrict__ feats,
    const int*   __restrict__ lengths,
    const float* __restrict__ lt,
    float*       __restrict__ part)
{
  const int lane = threadIdx.x;
  const int h    = lane >> 4;
  const int bl   = lane & 15;
  const int b    = blockIdx.x * 16 + bl;
  const int len  = lengths[b];

  // ---- constant A fragments: E^T[j,i] = exp(lt[i,j]) ----
  v2f A[3][12];
#pragma unroll
  for (int frag = 0; frag < 3; ++frag) {
    const int j = frag * 16 + bl;
#pragma unroll
    for (int c = 0; c < 12; ++c) {
      const int i0 = 4 * c + 2 * h;
      A[frag][c].x = __expf(lt[i0 * N_TAGS + j]);
      A[frag][c].y = __expf(lt[(i0 + 1) * N_TAGS + j]);
    }
  }

  // ---- alpha init: alpha_T[j,b] = lt[ROOT,j] + feats[b,0,j] ----
  float alpha[24];
  const size_t fbase = (size_t)b * MAXT * N_TAGS;
#pragma unroll
  for (int frag = 0; frag < 3; ++frag) {
    const int jb = frag * 16 + 8 * h;
    float4 f0 = *(const float4*)(feats + fbase + jb);
    float4 f1 = *(const float4*)(feats + fbase + jb + 4);
    float4 r0 = *(const float4*)(lt + ROOT_T * N_TAGS + jb);
    float4 r1 = *(const float4*)(lt + ROOT_T * N_TAGS + jb + 4);
    alpha[frag*8+0] = f0.x + r0.x;  alpha[frag*8+1] = f0.y + r0.y;
    alpha[frag*8+2] = f0.z + r0.z;  alpha[frag*8+3] = f0.w + r0.w;
    alpha[frag*8+4] = f1.x + r1.x;  alpha[frag*8+5] = f1.y + r1.y;
    alpha[frag*8+6] = f1.z + r1.z;  alpha[frag*8+7] = f1.w + r1.w;
  }

  for (int t = 1; t < MAXT; ++t) {
    // feats tile for this step, same C-layout as alpha (issued early: the
    // load latency hides under the max/exp/bpermute/WMMA chain below)
    float f[24];
    const float* fp = feats + fbase + (size_t)t * N_TAGS;
#pragma unroll
    for (int frag = 0; frag < 3; ++frag) {
      const int jb = frag * 16 + 8 * h;
      float4 f0 = *(const float4*)(fp + jb);
      float4 f1 = *(const float4*)(fp + jb + 4);
      f[frag*8+0] = f0.x; f[frag*8+1] = f0.y; f[frag*8+2] = f0.z; f[frag*8+3] = f0.w;
      f[frag*8+4] = f1.x; f[frag*8+5] = f1.y; f[frag*8+6] = f1.z; f[frag*8+7] = f1.w;
    }

    // per-batch-column max over all 48 tags: TREE reduction (depth ~5)
    float m12[12];
#pragma unroll
    for (int k = 0; k < 12; ++k) m12[k] = fmaxf(alpha[k], alpha[k + 12]);
    float m6[6];
#pragma unroll
    for (int k = 0; k < 6; ++k) m6[k] = fmaxf(m12[k], m12[k + 6]);
    float m3a = fmaxf(m6[0], m6[3]);
    float m3b = fmaxf(m6[1], m6[4]);
    float m3c = fmaxf(m6[2], m6[5]);
    float mx  = fmaxf(fmaxf(m3a, m3b), m3c);
    mx = fmaxf(mx, __shfl_xor(mx, 16, 32));

    float e[24];
#pragma unroll
    for (int k = 0; k < 24; ++k) e[k] = __expf(alpha[k] - mx);

    // Re-stripe e into WMMA B-layout: lane (h, bl) needs tag i = 4c + 2h + v.
    // Source of tag i: reg (i/16)*8 + (i&7), lane bl + 16*((i>>3)&1).
    float eb[24];
#pragma unroll
    for (int c = 0; c < 12; ++c) {
#pragma unroll
      for (int v = 0; v < 2; ++v) {
        const int iA = 4 * c + v;       // value needed by half 0
        const int iB = 4 * c + 2 + v;   // value needed by half 1
        float tA = __shfl(e[(iA >> 4) * 8 + (iA & 7)], bl + (((iA >> 3) & 1) << 4), 32);
        float tB = __shfl(e[(iB >> 4) * 8 + (iB & 7)], bl + (((iB >> 3) & 1) << 4), 32);
        eb[c * 2 + v] = h ? tB : tA;
      }
    }

    // S[j,b] = E^T @ exp(alpha - m) : 3 row fragments, each with TWO
    // independent accumulator chains (even/odd K-chunks) to halve the
    // WMMA RAW-chain depth, merged with one add pass.
    const bool upd = (t < len);
#pragma unroll
    for (int frag = 0; frag < 3; ++frag) {
      v8f acc0 = {0.f, 0.f, 0.f, 0.f, 0.f, 0.f, 0.f, 0.f};
      v8f acc1 = {0.f, 0.f, 0.f, 0.f, 0.f, 0.f, 0.f, 0.f};
#pragma unroll
      for (int c = 0; c < 12; c += 2) {
        v2f bv0; bv0.x = eb[2*c];     bv0.y = eb[2*c + 1];
        v2f bv1; bv1.x = eb[2*c + 2]; bv1.y = eb[2*c + 3];
        acc0 = __builtin_amdgcn_wmma_f32_16x16x4_f32(
            false, A[frag][c],     false, bv0, (short)0, acc0, false, false);
        acc1 = __builtin_amdgcn_wmma_f32_16x16x4_f32(
            false, A[frag][c + 1], false, bv1, (short)0, acc1, false, false);
      }
#pragma unroll
      for (int v = 0; v < 8; ++v) {
        float sv = acc0[v] + acc1[v];
        float nv = f[frag * 8 + v] + mx + __logf(sv);
        alpha[frag * 8 + v] = upd ? nv : alpha[frag * 8 + v];
      }
    }
  }

  // partition[b] = logsumexp_j(alpha[j,b] + lt[j, END])
  float p[24];
#pragma unroll
  for (int frag = 0; frag < 3; ++frag) {
#pragma unroll
    for (int v = 0; v < 8; ++v) {
      const int j = frag * 16 + 8 * h + v;
      p[frag * 8 + v] = alpha[frag * 8 + v] + lt[j * N_TAGS + END_T];
    }
  }
  float q12[12];
#pragma unroll
  for (int k = 0; k < 12; ++k) q12[k] = fmaxf(p[k], p[k + 12]);
  float q6[6];
#pragma unroll
  for (int k = 0; k < 6; ++k) q6[k] = fmaxf(q12[k], q12[k + 6]);
  float pm = fmaxf(fmaxf(fmaxf(q6[0], q6[3]), fmaxf(q6[1], q6[4])),
                   fmaxf(q6[2], q6[5]));
  pm = fmaxf(pm, __shfl_xor(pm, 16, 32));
  float s = 0.f;
#pragma unroll
  for (int k = 0; k < 24; ++k) s += __expf(p[k] - pm);
  s += __shfl_xor(s, 16, 32);
  if (h == 0) part[b] = __logf(s) + pm;
}

// ---------------------------------------------------------------------------
// Gold-path score (pure gathers) + final output.
// ---------------------------------------------------------------------------
__global__ void crf_gold(const float* __restrict__ feats,
                         const int*   __restrict__ tags,
                         const int*   __restrict__ lengths,
                         const float* __restrict__ lt,
                         const float* __restrict__ part,
                         float*       __restrict__ out)
{
  const int b = blockIdx.x * blockDim.x + threadIdx.x;
  if (b >= BATCH) return;
  const int len = lengths[b];                 // always >= 1
  const int*   tg = tags  + (size_t)b * MAXT;
  const float* fb = feats + (size_t)b * MAXT * N_TAGS;

  int tprev = tg[0];
  float gold = lt[ROOT_T * N_TAGS + tprev] + fb[tprev];   // t=0 emit + root trans
  for (int t = 1; t < len; ++t) {
    const int tc = tg[t];
    gold += lt[tprev * N_TAGS + tc];          // transition, masked by t < len
    gold += fb[(size_t)t * N_TAGS + tc];      // emission,   masked by t < len
    tprev = tc;
  }
  gold += lt[tprev * N_TAGS + END_T];         // tags[len-1] -> END
  out[b] = part[b] - gold;
}

extern "C" void kernel_launch(void* const* d_in, const int* in_sizes, int n_in,
                              void* d_out, int out_size, void* d_ws, size_t ws_size,
                              hipStream_t stream) {
  const float* feats   = (const float*)d_in[0];   // (512, 256, 48) f32
  const int*   tags    = (const int*)  d_in[1];   // (512, 256) i32
  const int*   lengths = (const int*)  d_in[2];   // (512,) i32
  const float* lt      = (const float*)d_in[3];   // (48, 48) f32
  float* part = (float*)d_ws;                     // 512 floats scratch
  float* out  = (float*)d_out;

  crf_forward<<<BATCH / 16, 32, 0, stream>>>(feats, lengths, lt, part);
  crf_gold<<<(BATCH + 127) / 128, 128, 0, stream>>>(feats, tags, lengths, lt, part, out);
}